// LGConv_layer_72688026518112
// MI455X (gfx1250) — compile-verified
//
#include <hip/hip_runtime.h>
#include <hip/hip_bf16.h>
#include <cmath>

#define H 128
#define NGRAPH 50
#define EPSILON 1e-5f

typedef __attribute__((ext_vector_type(2))) float v2f;
typedef __attribute__((ext_vector_type(8))) float v8f;

static __device__ inline v8f vzero8() {
  v8f v;
#pragma unroll
  for (int i = 0; i < 8; ++i) v[i] = 0.f;
  return v;
}

// ---------------- zero workspace region ----------------
__global__ void k_zero(float* __restrict__ p, long n) {
  long i = (long)blockIdx.x * blockDim.x + threadIdx.x;
  if (i < n) p[i] = 0.f;
}

// ---------------- weighted degree by target ----------------
__global__ void k_deg(const int* __restrict__ col, const float* __restrict__ ea,
                      float* __restrict__ deg, int E) {
  int e = blockIdx.x * blockDim.x + threadIdx.x;
  if (e < E) atomicAdd(deg + col[e], ea[e]);
}

// ---------------- dinv = deg > 0 ? deg^-0.5 : 0  (in place) ----------------
__global__ void k_dinv(float* __restrict__ deg, int N) {
  int i = blockIdx.x * blockDim.x + threadIdx.x;
  if (i < N) {
    float d = deg[i];
    deg[i] = (d > 0.f) ? (1.f / sqrtf(d)) : 0.f;
  }
}

// ---------------- per-graph node counts ----------------
__global__ void k_cnt(const int* __restrict__ bp, float* __restrict__ gcnt, int N) {
  int i = blockIdx.x * blockDim.x + threadIdx.x;
  if (i < N) atomicAdd(gcnt + bp[i], 1.f);
}

// ---------------- edge scatter: hbuf[col] += node[row] * w  (one wave per edge) ----------------
__global__ void k_scatter(const int* __restrict__ row, const int* __restrict__ col,
                          const float* __restrict__ ea, const float* __restrict__ dinv,
                          const float* __restrict__ node, float* __restrict__ hbuf, int E) {
  const int lane = threadIdx.x & 31;
  const int e = blockIdx.x * (blockDim.x >> 5) + (threadIdx.x >> 5);
  if (e >= E) return;
  const int r = row[e], c = col[e];
  const float w = dinv[r] * ea[e] * dinv[c];
  const float4 v = ((const float4*)(node + (long)r * H))[lane];
  float* dst = hbuf + (long)c * H + lane * 4;
  atomicAdd(dst + 0, v.x * w);
  atomicAdd(dst + 1, v.y * w);
  atomicAdd(dst + 2, v.z * w);
  atomicAdd(dst + 3, v.w * w);
}

// ---------------- fused: h = node + agg (written back) + per-graph sum / sumsq via WMMA ----------------
// Each block owns 256 contiguous nodes; wave w owns feature group [16w, 16w+16).
// Uniform 4-node runs (same graph) are reduced with V_WMMA_F32_16X16X4_F32 using an
// all-ones A matrix: every lane's c[0] = column-sum of its feature (N = lane%16),
// invariant to the K-lane ordering of B. Boundary groups take a scalar-atomic fallback.
__global__ void k_moments(const float* __restrict__ node, float* __restrict__ hbuf,
                          const int* __restrict__ bp, float* __restrict__ gsum,
                          float* __restrict__ gsumsq, int N) {
  const int lane = threadIdx.x & 31;
  const int wave = threadIdx.x >> 5;
  const int fbase = wave * 16;
  const int fl = lane & 15;     // feature within group (== N index of this lane)
  const int kk = lane >> 4;     // node sub-index 0/1 for element 0 of the pair

  const int n0 = blockIdx.x * 256;
  const int nEnd = min(N, n0 + 256);

  v8f cS = vzero8();
  v8f cQ = vzero8();
  v2f ones; ones[0] = 1.f; ones[1] = 1.f;
  int curg = -1;

  for (int n = n0; n < nEnd; n += 4) {
    const bool full = (n + 4 <= nEnd);
    const int ga = bp[n];
    const int gb = bp[min(n + 3, nEnd - 1)];
    if (full && ga == gb) {                 // wave-uniform => EXEC all-1s at WMMA
      if (ga != curg) {
        if (curg >= 0) {
          if (lane < 16) atomicAdd(gsum  + curg * H + fbase + fl, cS[0]);
          else           atomicAdd(gsumsq + curg * H + fbase + fl, cQ[0]);
        }
        cS = vzero8();
        cQ = vzero8();
        curg = ga;
      }
      const long i0 = (long)(n + kk) * H + fbase + fl;
      const long i1 = (long)(n + kk + 2) * H + fbase + fl;
      const float h0 = node[i0] + hbuf[i0];
      const float h1 = node[i1] + hbuf[i1];
      hbuf[i0] = h0;
      hbuf[i1] = h1;
      v2f B;  B[0] = h0;       B[1] = h1;
      v2f B2; B2[0] = h0 * h0; B2[1] = h1 * h1;
      cS = __builtin_amdgcn_wmma_f32_16x16x4_f32(false, ones, false, B,  (short)0, cS, false, false);
      cQ = __builtin_amdgcn_wmma_f32_16x16x4_f32(false, ones, false, B2, (short)0, cQ, false, false);
    } else {                                // graph boundary / tail: scalar fallback
#pragma unroll
      for (int j = 0; j < 2; ++j) {
        const int nk = n + kk + 2 * j;
        if (nk < nEnd) {
          const long idx = (long)nk * H + fbase + fl;
          const int g = bp[nk];
          const float hv = node[idx] + hbuf[idx];
          hbuf[idx] = hv;
          atomicAdd(gsum  + g * H + fbase + fl, hv);
          atomicAdd(gsumsq + g * H + fbase + fl, hv * hv);
        }
      }
    }
  }
  if (curg >= 0) {
    if (lane < 16) atomicAdd(gsum  + curg * H + fbase + fl, cS[0]);
    else           atomicAdd(gsumsq + curg * H + fbase + fl, cQ[0]);
  }
}

// ---------------- per-(graph,feature) stats: mean*scale and weight*rstd ----------------
__global__ void k_finalize(const float* __restrict__ gsum, const float* __restrict__ gsumsq,
                           const float* __restrict__ gcnt, const float* __restrict__ weight,
                           const float* __restrict__ mscale, float* __restrict__ gmean_s,
                           float* __restrict__ grstd_w) {
  int i = blockIdx.x * blockDim.x + threadIdx.x;
  if (i >= NGRAPH * H) return;
  const int g = i / H, f = i % H;
  const float cnt = fmaxf(gcnt[g], 1.f);
  const float mean = gsum[i] / cnt;
  const float msq  = gsumsq[i] / cnt;
  const float ms = mean * mscale[f];
  float var = msq - 2.f * ms * mean + ms * ms;   // E[(h-ms)^2]
  var = fmaxf(var, 0.f);
  gmean_s[i] = ms;
  grstd_w[i] = weight[f] / sqrtf(var + EPSILON);
}

// ---------------- normalize + bias + ReLU ----------------
__global__ void k_out(const float* __restrict__ hbuf, const int* __restrict__ bp,
                      const float* __restrict__ gmean_s, const float* __restrict__ grstd_w,
                      const float* __restrict__ bias, float* __restrict__ out, int N) {
  long i = (long)blockIdx.x * blockDim.x + threadIdx.x;
  if (i >= (long)N * H) return;
  const int n = (int)(i >> 7);
  const int f = (int)(i & 127);
  const int g = bp[n];
  const float c = hbuf[i] - gmean_s[g * H + f];
  out[i] = fmaxf(c * grstd_w[g * H + f] + bias[f], 0.f);
}

extern "C" void kernel_launch(void* const* d_in, const int* in_sizes, int n_in,
                              void* d_out, int out_size, void* d_ws, size_t ws_size,
                              hipStream_t stream) {
  const float* node   = (const float*)d_in[0];
  const int*   eidx   = (const int*)  d_in[1];
  const float* ea     = (const float*)d_in[2];
  const int*   bp     = (const int*)  d_in[3];
  const float* weight = (const float*)d_in[4];
  const float* bias   = (const float*)d_in[5];
  const float* mscale = (const float*)d_in[6];
  float* out = (float*)d_out;

  const int N = in_sizes[0] / H;
  const int E = in_sizes[2];
  const int* row = eidx;
  const int* col = eidx + E;

  float* ws = (float*)d_ws;
  float* deg    = ws;                          // N   (becomes dinv in place)
  float* hbuf   = deg + N;                     // N*H (agg, then h in place)
  float* gsum   = hbuf + (long)N * H;          // G*H
  float* gsumsq = gsum + NGRAPH * H;           // G*H
  float* gcnt   = gsumsq + NGRAPH * H;         // G
  float* gmean  = gcnt + NGRAPH;               // G*H
  float* grstd  = gmean + NGRAPH * H;          // G*H

  const long zn = (long)N + (long)N * H + 2L * NGRAPH * H + NGRAPH; // deg..gcnt contiguous
  k_zero<<<(unsigned)((zn + 255) / 256), 256, 0, stream>>>(deg, zn);
  k_deg<<<(E + 255) / 256, 256, 0, stream>>>(col, ea, deg, E);
  k_dinv<<<(N + 255) / 256, 256, 0, stream>>>(deg, N);
  k_cnt<<<(N + 255) / 256, 256, 0, stream>>>(bp, gcnt, N);
  k_scatter<<<(E + 7) / 8, 256, 0, stream>>>(row, col, ea, deg, node, hbuf, E);
  k_moments<<<(N + 255) / 256, 256, 0, stream>>>(node, hbuf, bp, gsum, gsumsq, N);
  k_finalize<<<(NGRAPH * H + 255) / 256, 256, 0, stream>>>(gsum, gsumsq, gcnt, weight, mscale, gmean, grstd);
  k_out<<<(unsigned)(((long)N * H + 255) / 256), 256, 0, stream>>>(hbuf, bp, gmean, grstd, bias, out, N);
}